// GAE_5377299054952
// MI455X (gfx1250) — compile-verified
//
#include <hip/hip_runtime.h>

typedef __attribute__((ext_vector_type(2))) float v2f;
typedef __attribute__((ext_vector_type(8))) float v8f;

#define N_NODES 50000
#define K_DIM   256   // inner dim for both layers (d_in = d_h = 256)

// ---------------------------------------------------------------------------
// Degree / normalization
// ---------------------------------------------------------------------------
__global__ void zero_u32_kernel(unsigned* __restrict__ p, int n) {
    int i = blockIdx.x * blockDim.x + threadIdx.x;
    if (i < n) p[i] = 0u;
}

__global__ void count_deg_kernel(const int* __restrict__ dst, unsigned* __restrict__ deg, int E) {
    int i = blockIdx.x * blockDim.x + threadIdx.x;
    if (i < E) atomicAdd(&deg[dst[i]], 1u);
}

__global__ void make_dis_kernel(const unsigned* __restrict__ deg, float* __restrict__ dis, int n) {
    int i = blockIdx.x * blockDim.x + threadIdx.x;
    if (i < n) dis[i] = rsqrtf((float)(deg[i] + 1u));  // +1 = self loop, so deg>0 always
}

// ---------------------------------------------------------------------------
// fp32 WMMA GEMM:  H[nrows x NOUT] = act(A[nrows x 256]) @ W[256 x NOUT]
// Block = 256 threads = 8 waves. Block tile = 512(M) x 16(N).
// Each wave: 64x16 tile = FOUR independent 16x16 accumulators sharing one
// B fragment per k-step -> 4 parallel WMMA chains (no C-RAW serialization),
// ds traffic amortized 4x. W panel (256x16 = 16 KB) staged in LDS once.
//
// A-fragment layout (ISA 7.12.2, 32-bit A 16x4): lanes 0-15 hold K=0,1 in
// v0,v1; lanes 16-31 hold K=2,3.  B (4x16): lanes 0-15 = N, v0/v1 = K0/K1,
// lanes 16-31 = K2/K3.  C/D: VGPR r = row r (lanes 0-15) / row r+8 (16-31).
// ---------------------------------------------------------------------------
template <int NOUT, bool RELU_IN>
__global__ __launch_bounds__(256) void gemm_wmma_f32_kernel(
    const float* __restrict__ A, const float* __restrict__ W,
    float* __restrict__ H, int nrows) {
    __shared__ float lds_w[K_DIM * 16];

    const int tid = threadIdx.x;
    const int n0  = blockIdx.x * 16;

    // Stage the 256x16 W panel into LDS (coalesced: 16 consecutive cols/row).
    #pragma unroll
    for (int i = tid; i < K_DIM * 16; i += 256)
        lds_w[i] = W[(i >> 4) * NOUT + n0 + (i & 15)];
    __syncthreads();

    const int wave = tid >> 5;
    const int lane = tid & 31;
    const int m0   = blockIdx.y * 512 + wave * 64;   // 64 rows per wave
    if (m0 >= nrows) return;                 // wave-uniform exit; EXEC stays all-1s

    const int kh    = (lane >> 4) * 2;       // 0 for lanes 0-15, 2 for lanes 16-31
    const int ncol  = lane & 15;
    const int rbase = m0 + (lane & 15);
    const int rmax  = nrows - 1;

    // Per-lane row clamp for the tail block (lane-level select, EXEC untouched).
    const float* arow0 = A + (long)min(rbase +  0, rmax) * K_DIM + kh;
    const float* arow1 = A + (long)min(rbase + 16, rmax) * K_DIM + kh;
    const float* arow2 = A + (long)min(rbase + 32, rmax) * K_DIM + kh;
    const float* arow3 = A + (long)min(rbase + 48, rmax) * K_DIM + kh;
    const float* bcol  = lds_w + kh * 16 + ncol;

    v8f c0 = {}, c1 = {}, c2 = {}, c3 = {};
    #pragma unroll 4
    for (int k = 0; k < K_DIM; k += 4) {
        v2f b;
        b.x = bcol[k * 16];                              // B[k+kh  ][ncol]
        b.y = bcol[k * 16 + 16];                         // B[k+kh+1][ncol]
        v2f a0 = *(const v2f*)(arow0 + k);
        v2f a1 = *(const v2f*)(arow1 + k);
        v2f a2 = *(const v2f*)(arow2 + k);
        v2f a3 = *(const v2f*)(arow3 + k);
        if (RELU_IN) {
            a0.x = fmaxf(a0.x, 0.0f); a0.y = fmaxf(a0.y, 0.0f);
            a1.x = fmaxf(a1.x, 0.0f); a1.y = fmaxf(a1.y, 0.0f);
            a2.x = fmaxf(a2.x, 0.0f); a2.y = fmaxf(a2.y, 0.0f);
            a3.x = fmaxf(a3.x, 0.0f); a3.y = fmaxf(a3.y, 0.0f);
        }
        c0 = __builtin_amdgcn_wmma_f32_16x16x4_f32(false, a0, false, b, (short)0, c0, false, false);
        c1 = __builtin_amdgcn_wmma_f32_16x16x4_f32(false, a1, false, b, (short)0, c1, false, false);
        c2 = __builtin_amdgcn_wmma_f32_16x16x4_f32(false, a2, false, b, (short)0, c2, false, false);
        c3 = __builtin_amdgcn_wmma_f32_16x16x4_f32(false, a3, false, b, (short)0, c3, false, false);
    }

    // Store: sub-tile t covers rows m0+16t .. m0+16t+15 (16-aligned; nrows is
    // a multiple of 16, so the guard is wave-uniform per sub-tile).
    const long colofs = (long)(((lane >> 4) << 3)) * NOUT + n0 + ncol;
    {
        float* hrow = H + (long)(m0 + 0) * NOUT + colofs;
        #pragma unroll
        for (int r = 0; r < 8; ++r) hrow[(long)r * NOUT] = c0[r];
    }
    if (m0 + 16 < nrows) {
        float* hrow = H + (long)(m0 + 16) * NOUT + colofs;
        #pragma unroll
        for (int r = 0; r < 8; ++r) hrow[(long)r * NOUT] = c1[r];
    }
    if (m0 + 32 < nrows) {
        float* hrow = H + (long)(m0 + 32) * NOUT + colofs;
        #pragma unroll
        for (int r = 0; r < 8; ++r) hrow[(long)r * NOUT] = c2[r];
    }
    if (m0 + 48 < nrows) {
        float* hrow = H + (long)(m0 + 48) * NOUT + colofs;
        #pragma unroll
        for (int r = 0; r < 8; ++r) hrow[(long)r * NOUT] = c3[r];
    }
}

// ---------------------------------------------------------------------------
// agg[n][c] = h[n][c] * dis[n]^2 + bias[c]   (self-loop term + bias, fused)
// ---------------------------------------------------------------------------
template <int C>
__global__ void init_agg_kernel(const float* __restrict__ h, const float* __restrict__ dis,
                                const float* __restrict__ bias, float* __restrict__ agg, int n) {
    int i = blockIdx.x * blockDim.x + threadIdx.x;
    if (i < n * C) {
        int node = i / C;
        int c    = i - node * C;
        float d  = dis[node];
        agg[i]   = h[i] * d * d + bias[c];
    }
}

// ---------------------------------------------------------------------------
// Edge scatter: agg[dst] += h[src] * dis[src]*dis[dst]
// Consecutive lanes = consecutive channel chunks of one edge -> coalesced
// float4 gathers; 4x global_atomic_add_f32 (L2-resident, 51 MB << 192 MB L2).
// ---------------------------------------------------------------------------
template <int C>
__global__ __launch_bounds__(256) void scatter_edges_kernel(
    const int* __restrict__ src, const int* __restrict__ dst,
    const float* __restrict__ h, const float* __restrict__ dis,
    float* __restrict__ agg, int E) {
    const int  per   = C / 4;
    const long idx   = (long)blockIdx.x * blockDim.x + threadIdx.x;
    const long total = (long)E * per;
    if (idx >= total) return;
    const int e  = (int)(idx / per);
    const int c4 = (int)(idx - (long)e * per);

    const int   s   = src[e];
    const int   d   = dst[e];
    const float nrm = dis[s] * dis[d];

    const float4 v   = *(const float4*)(h + (long)s * C + c4 * 4);
    float*       out = agg + (long)d * C + c4 * 4;
    atomicAdd(out + 0, v.x * nrm);
    atomicAdd(out + 1, v.y * nrm);
    atomicAdd(out + 2, v.z * nrm);
    atomicAdd(out + 3, v.w * nrm);
}

// ---------------------------------------------------------------------------
// Decode: y[e] = dot(z[i[e]], z[j[e]]) over 128 dims. One wave32 per edge:
// float4 per lane + shfl_xor tree reduction.
// ---------------------------------------------------------------------------
__global__ __launch_bounds__(256) void decode_dot_kernel(
    const int* __restrict__ eli, const float* __restrict__ z,
    float* __restrict__ y, int EL) {
    const int gw   = (int)(((long)blockIdx.x * blockDim.x + threadIdx.x) >> 5);
    const int lane = threadIdx.x & 31;
    if (gw >= EL) return;                    // wave-uniform

    const int a = eli[gw];
    const int b = eli[EL + gw];
    const float4 va = *(const float4*)(z + (long)a * 128 + lane * 4);
    const float4 vb = *(const float4*)(z + (long)b * 128 + lane * 4);
    float p = va.x * vb.x + va.y * vb.y + va.z * vb.z + va.w * vb.w;
    #pragma unroll
    for (int off = 16; off > 0; off >>= 1)
        p += __shfl_xor(p, off);
    if (lane == 0) y[gw] = p;
}

// ---------------------------------------------------------------------------
extern "C" void kernel_launch(void* const* d_in, const int* in_sizes, int n_in,
                              void* d_out, int out_size, void* d_ws, size_t ws_size,
                              hipStream_t stream) {
    const float* x   = (const float*)d_in[0];
    const int*   ei  = (const int*)d_in[1];   // [2][E] flat: src then dst
    const int*   eli = (const int*)d_in[2];   // [2][EL]
    const float* W1  = (const float*)d_in[3];
    const float* b1  = (const float*)d_in[4];
    const float* W2  = (const float*)d_in[5];
    const float* b2  = (const float*)d_in[6];
    float*       y   = (float*)d_out;

    const int N  = N_NODES;
    const int E  = in_sizes[1] / 2;
    const int EL = in_sizes[2] / 2;

    // Workspace layout (~103 MB): deg | dis | h (50000x256) | agg (50000x256).
    // Layer 2 reuses: h2 -> h buffer, z2 -> agg buffer.
    char* ws = (char*)d_ws;
    unsigned* deg = (unsigned*)ws;  ws += ((size_t)N * 4 + 255) & ~(size_t)255;
    float*    dis = (float*)ws;     ws += ((size_t)N * 4 + 255) & ~(size_t)255;
    float*    h   = (float*)ws;     ws += (size_t)N * 256 * 4;
    float*    agg = (float*)ws;

    // ---- normalization: deg (in-degree) -> dis = rsqrt(deg+1) ----
    zero_u32_kernel<<<(N + 255) / 256, 256, 0, stream>>>(deg, N);
    count_deg_kernel<<<(E + 255) / 256, 256, 0, stream>>>(ei + E, deg, E);
    make_dis_kernel<<<(N + 255) / 256, 256, 0, stream>>>(deg, dis, N);

    const int mblocks = (N + 511) / 512;     // 512 rows per block (8 waves x 64)

    // ---- layer 1: h = x @ W1 ; agg = h*dis^2 + b1 ; scatter edges ----
    gemm_wmma_f32_kernel<256, false><<<dim3(16, mblocks), 256, 0, stream>>>(x, W1, h, N);
    init_agg_kernel<256><<<((N * 256) + 255) / 256, 256, 0, stream>>>(h, dis, b1, agg, N);
    {
        const long total = (long)E * 64;
        scatter_edges_kernel<256><<<(unsigned)((total + 255) / 256), 256, 0, stream>>>(
            ei, ei + E, h, dis, agg, E);
    }

    // ---- layer 2: h2 = relu(agg) @ W2 (ReLU fused into A load) ----
    gemm_wmma_f32_kernel<128, true><<<dim3(8, mblocks), 256, 0, stream>>>(agg, W2, h, N);
    float* z2 = agg;  // reuse (agg1 dead after GEMM2)
    init_agg_kernel<128><<<((N * 128) + 255) / 256, 256, 0, stream>>>(h, dis, b2, z2, N);
    {
        const long total = (long)E * 32;
        scatter_edges_kernel<128><<<(unsigned)((total + 255) / 256), 256, 0, stream>>>(
            ei, ei + E, h, dis, z2, E);
    }

    // ---- decode: edge dot products ----
    decode_dot_kernel<<<((long)EL * 32 + 255) / 256, 256, 0, stream>>>(eli, z2, y, EL);
}